// classic_slab1D_eqx_87617332838686
// MI455X (gfx1250) — compile-verified
//
#include <hip/hip_runtime.h>
#include <math.h>

// ---------------------------------------------------------------------------
// classic slab 1D ocean model: U_{t+1} = a*U_t + b_t  (complex, NT steps)
// Decay-truncated block scan + WMMA triangular carry combine (gfx1250).
// ---------------------------------------------------------------------------

typedef __attribute__((ext_vector_type(2))) float v2f;
typedef __attribute__((ext_vector_type(8))) float v8f;

#define NTQ   5040000     // model steps
#define NSUBQ 60          // substeps per forcing sample
#define NFQ   84000       // forcing length
#define DTQ   60.0f
#define FCQ   1.0e-4f

#define LCH       128     // steps per lane
#define THREADS   256
#define WARM      12288   // warm-up steps (96 lanes): |a|^WARM < 3e-11
#define SPAN      20480   // output steps per block (160 lanes)
#define WARMLANES (WARM / LCH)                     // 96
#define NBLK      ((NTQ - 1 + SPAN - 1) / SPAN)    // 247
#define FSTAGE    576     // staged forcing samples (need <= 549)

#if defined(__gfx1250__) && \
    __has_builtin(__builtin_amdgcn_global_load_async_to_lds_b32) && \
    __has_builtin(__builtin_amdgcn_s_wait_asynccnt)
#define SLAB_ASYNC 1
typedef __attribute__((address_space(1))) int* gint_p;
typedef __attribute__((address_space(3))) int* lint_p;
#else
#define SLAB_ASYNC 0
#endif

// One 128-step lane chunk. Forcing is piecewise-linear over 60-step segments,
// so hoist LDS reads + slope per segment and advance b incrementally:
// inner loop = 4 FMA (complex step) + 2 adds, no LDS, no index math.
template <bool WRITE>
__device__ __forceinline__ void run_chunk(int t0, int itf0, int rr0, int base_f,
                                          float ar, float ai, float c0,
                                          const float* fx, const float* fy,
                                          float& ur, float& ui,
                                          float* __restrict__ out)
{
    int t = t0, itf = itf0, r = rr0;
    int remaining = LCH;
    while (remaining > 0) {
        int seg = NSUBQ - r;
        if (seg > remaining) seg = remaining;
        float br, bi, dbr, dbi;
        if (t < 0) {                      // warm-up before t=0: zero forcing
            br = bi = dbr = dbi = 0.0f;   // (t=0 is a segment boundary)
        } else {
            int li  = itf - base_f;
            int i2  = itf + 1; if (i2 > NFQ - 1) i2 = NFQ - 1;  // JAX clamp
            int li2 = i2 - base_f;
            float f1r = fx[li], f2r = fx[li2];
            float f1i = fy[li], f2i = fy[li2];
            float sr = (f2r - f1r) * (1.0f / NSUBQ);
            float si = (f2i - f1i) * (1.0f / NSUBQ);
            br  = c0 * fmaf((float)r, sr, f1r);
            bi  = c0 * fmaf((float)r, si, f1i);
            dbr = c0 * sr;
            dbi = c0 * si;
        }
        #pragma unroll 4
        for (int s = 0; s < seg; ++s) {
            float nur = fmaf(ar, ur, fmaf(-ai, ui, br));
            float nui = fmaf(ar, ui, fmaf(ai, ur, bi));
            ur = nur; ui = nui;
            if (WRITE) {
                if (t <= NTQ - 2) {       // out[t+1] = U_{t+1}
                    __builtin_nontemporal_store(ur, out + (t + 1));
                    __builtin_nontemporal_store(ui, out + (NTQ + t + 1));
                }
            }
            br += dbr; bi += dbi;
            ++t;
        }
        r = 0; ++itf;
        remaining -= seg;
    }
}

__global__ __launch_bounds__(THREADS)
void slab_scan_kernel(const float* __restrict__ pk,
                      const float* __restrict__ TAx,
                      const float* __restrict__ TAy,
                      float* __restrict__ out)
{
    __shared__ float fx[FSTAGE], fy[FSTAGE];       // staged forcing
    __shared__ float carR[THREADS], carI[THREADS]; // per-lane local scan finals
    __shared__ float YwR[256], YwI[256];           // 16x16 in-group scans
    __shared__ float ApR[17], ApI[17];             // A^0..A^16, A = a^128

    const int tid = threadIdx.x;
    const int blk = blockIdx.x;
    const int blockStart = blk * SPAN - WARM;

    // uniform model constants
    const float K0 = __expf(pk[0]);
    const float K1 = __expf(pk[1]);
    const float ar = 1.0f - DTQ * K1;   // Re(a)
    const float ai = -DTQ * FCQ;        // Im(a)
    const float c0 = DTQ * K0;          // forcing scale

    // ---- stage forcing slice into LDS (async global->LDS when available) ----
    const int base_f = (blockStart > 0 ? blockStart : 0) / NSUBQ;
#if SLAB_ASYNC
    for (int i = tid; i < FSTAGE; i += THREADS) {
        int idx = base_f + i;
        if (idx > NFQ - 1) idx = NFQ - 1;
        __builtin_amdgcn_global_load_async_to_lds_b32(
            (gint_p)(TAx + idx), (lint_p)(&fx[i]), 0, 0);
        __builtin_amdgcn_global_load_async_to_lds_b32(
            (gint_p)(TAy + idx), (lint_p)(&fy[i]), 0, 0);
    }
#else
    for (int i = tid; i < FSTAGE; i += THREADS) {
        int idx = base_f + i;
        if (idx > NFQ - 1) idx = NFQ - 1;
        fx[i] = TAx[idx];
        fy[i] = TAy[idx];
    }
#endif

    // ---- thread 0: powers of A = a^128 (7 squarings), A^0..A^16 ----
    if (tid == 0) {
        float Ar_ = ar, Ai_ = ai;
        for (int s = 0; s < 7; ++s) {
            float nr = Ar_ * Ar_ - Ai_ * Ai_;
            float ni = 2.0f * Ar_ * Ai_;
            Ar_ = nr; Ai_ = ni;
        }
        float pr = 1.0f, pi = 0.0f;
        for (int j = 0; j <= 16; ++j) {
            ApR[j] = pr; ApI[j] = pi;
            float nr = pr * Ar_ - pi * Ai_;
            float ni = pr * Ai_ + pi * Ar_;
            pr = nr; pi = ni;
        }
    }

    // ---- per-lane step bookkeeping ----
    const int t0 = blockStart + tid * LCH;
    int itf0, rr0;
    if (t0 >= 0) { itf0 = t0 / NSUBQ; rr0 = t0 - itf0 * NSUBQ; }
    else {
        int q = (-t0 + NSUBQ - 1) / NSUBQ;   // floor division for negatives
        itf0 = -q; rr0 = t0 + q * NSUBQ;
    }

#if SLAB_ASYNC
    __builtin_amdgcn_s_wait_asynccnt(0);     // own async loads landed in LDS
#endif
    __syncthreads();                          // all waves' staging + Apow ready

    // =========================== PASS 1: local scan ==========================
    {
        float ur = 0.0f, ui = 0.0f;
        run_chunk<false>(t0, itf0, rr0, base_f, ar, ai, c0, fx, fy, ur, ui, out);
        carR[tid] = ur; carI[tid] = ui;
    }
    __syncthreads();

    // ============ carry combine: Yw[m][g] = sum_{j<=m} A^(m-j) y[g*16+j] =====
    // Lower-triangular complex matmul on the matrix core, wave 0 only.
    if (tid < 32) {
        const int  lane = tid;
        const bool hi   = lane >= 16;
        const int  half = hi ? lane - 16 : lane;   // A row M / B,D column N
        v8f Dr = {}, Di = {};
        #pragma unroll
        for (int kb = 0; kb < 4; ++kb) {
            // A-matrix 16x4 f32 layout: VGPR0 -> K {0|2}, VGPR1 -> K {1|3}
            const int jx = kb * 4 + (hi ? 2 : 0);
            const int jy = jx + 1;
            const int dx = half - jx, dy = half - jy;
            const int cx = dx < 0 ? 0 : dx, cy = dy < 0 ? 0 : dy;
            const float mx = dx < 0 ? 0.0f : 1.0f;
            const float my = dy < 0 ? 0.0f : 1.0f;
            v2f ArT, AiT, nAiT, BrT, BiT;
            ArT.x  = ApR[cx] * mx;  ArT.y  = ApR[cy] * my;
            AiT.x  = ApI[cx] * mx;  AiT.y  = ApI[cy] * my;
            nAiT.x = -AiT.x;        nAiT.y = -AiT.y;
            // B-matrix 4x16: row K striped across lane halves, col N = half
            BrT.x = carR[half * 16 + jx]; BrT.y = carR[half * 16 + jy];
            BiT.x = carI[half * 16 + jx]; BiT.y = carI[half * 16 + jy];
            // complex product: Dr = Tr*Br - Ti*Bi ; Di = Ti*Br + Tr*Bi
            Dr = __builtin_amdgcn_wmma_f32_16x16x4_f32(false, ArT,  false, BrT,
                                                       (short)0, Dr, false, false);
            Dr = __builtin_amdgcn_wmma_f32_16x16x4_f32(false, nAiT, false, BiT,
                                                       (short)0, Dr, false, false);
            Di = __builtin_amdgcn_wmma_f32_16x16x4_f32(false, AiT,  false, BrT,
                                                       (short)0, Di, false, false);
            Di = __builtin_amdgcn_wmma_f32_16x16x4_f32(false, ArT,  false, BiT,
                                                       (short)0, Di, false, false);
        }
        // C/D layout: VGPR r -> M = r (lanes 0-15) / r+8 (lanes 16-31), N = half
        #pragma unroll
        for (int r = 0; r < 8; ++r) {
            const int row = hi ? r + 8 : r;
            YwR[row * 16 + half] = Dr[r];
            YwI[row * 16 + half] = Di[r];
        }
    }
    __syncthreads();

    // ---- exclusive carry into each lane: group Horner + in-group prefix ----
    const int m = tid & 15, g = tid >> 4;
    const float A16r = ApR[16], A16i = ApI[16];
    float Pr = 0.0f, Pi = 0.0f;             // sum_{h<g} A16^h * S_h
    for (int h = g - 1; h >= 0; --h) {
        float sr = YwR[15 * 16 + h], si = YwI[15 * 16 + h];
        float nr = fmaf(A16r, Pr, fmaf(-A16i, Pi, sr));
        float ni = fmaf(A16r, Pi, fmaf( A16i, Pr, si));
        Pr = nr; Pi = ni;
    }
    float initR, initI;
    if (m == 0) { initR = Pr; initI = Pi; }
    else {
        float amr = ApR[m], ami = ApI[m];
        float yr = YwR[(m - 1) * 16 + g], yi = YwI[(m - 1) * 16 + g];
        initR = fmaf(amr, Pr, fmaf(-ami, Pi, yr));
        initI = fmaf(amr, Pi, fmaf( ami, Pr, yi));
    }

    // ====================== PASS 2: rerun + stream outputs ===================
    if (blk == 0 && tid == 0) {             // U[0] = 0
        __builtin_nontemporal_store(0.0f, out);
        __builtin_nontemporal_store(0.0f, out + NTQ);
    }
    if (tid >= WARMLANES) {                 // only lanes inside the output span
        float ur = initR, ui = initI;
        run_chunk<true>(t0, itf0, rr0, base_f, ar, ai, c0, fx, fy, ur, ui, out);
    }
}

extern "C" void kernel_launch(void* const* d_in, const int* in_sizes, int n_in,
                              void* d_out, int out_size, void* d_ws, size_t ws_size,
                              hipStream_t stream) {
    (void)in_sizes; (void)n_in; (void)out_size; (void)d_ws; (void)ws_size;
    const float* pk  = (const float*)d_in[0];
    const float* TAx = (const float*)d_in[1];
    const float* TAy = (const float*)d_in[2];
    float* out = (float*)d_out;
    slab_scan_kernel<<<NBLK, THREADS, 0, stream>>>(pk, TAx, TAy, out);
}